// LIF_BTN_27848567947918
// MI455X (gfx1250) — compile-verified
//
#include <hip/hip_runtime.h>

// LIF scan over time, time-major x:[T,B,N] f32.
//   m = tau*v + x[t];  y = (m >= v_th);  v' = m*(1-y)  (hard reset)
// Pure streaming kernel: 128 MiB in + 128 MiB out -> HBM-bound (~11.5 us @ 23.3 TB/s).
// One thread owns 4 consecutive neurons; v lives in VGPRs across all T steps.

#define LIF_T   32
#define LIF_TAU 0.5f
#define LIF_VTH 1.0f

// Native clang vector type: required by __builtin_nontemporal_* and lowers to
// b128 vmem ops (HIP's float4 is a class and is rejected by the builtin).
typedef float v4f __attribute__((ext_vector_type(4)));

__global__ __launch_bounds__(256) void lif_btn_kernel(
    const v4f* __restrict__ x,   // [T, BN/4]
    v4f* __restrict__ y,         // [T, BN/4]
    int bn4)                     // BN/4 (vector columns)
{
    const int idx = blockIdx.x * blockDim.x + threadIdx.x;
    if (idx >= bn4) return;

    v4f v = (v4f)(0.0f);

    // Prime the prefetcher with the first planes (gfx1250 global_prefetch_b8).
    __builtin_prefetch(&x[idx], 0, 1);
    __builtin_prefetch(&x[(size_t)bn4 + idx], 0, 1);

#pragma unroll
    for (int t = 0; t < LIF_T; ++t) {
        const size_t off = (size_t)t * (size_t)bn4 + (size_t)idx;

        // Prefetch two timestep-planes ahead (4 MiB stride between planes).
        if (t + 2 < LIF_T) {
            __builtin_prefetch(&x[off + 2u * (size_t)bn4], 0, 1);
        }

        // Touch-once stream: non-temporal load -> global_load_b128 th:NT.
        const v4f xt = __builtin_nontemporal_load(&x[off]);

        v4f m;
        m.x = __builtin_fmaf(LIF_TAU, v.x, xt.x);
        m.y = __builtin_fmaf(LIF_TAU, v.y, xt.y);
        m.z = __builtin_fmaf(LIF_TAU, v.z, xt.z);
        m.w = __builtin_fmaf(LIF_TAU, v.w, xt.w);

        // Spike + hard reset, branch-free (v_cmp + v_cndmask).
        const bool fx = (m.x >= LIF_VTH);
        const bool fy = (m.y >= LIF_VTH);
        const bool fz = (m.z >= LIF_VTH);
        const bool fw = (m.w >= LIF_VTH);

        v4f yo;
        yo.x = fx ? 1.0f : 0.0f;
        yo.y = fy ? 1.0f : 0.0f;
        yo.z = fz ? 1.0f : 0.0f;
        yo.w = fw ? 1.0f : 0.0f;

        v.x = fx ? 0.0f : m.x;   // m*(1-y) with y in {0,1}
        v.y = fy ? 0.0f : m.y;
        v.z = fz ? 0.0f : m.z;
        v.w = fw ? 0.0f : m.w;

        // Touch-once output: non-temporal store -> global_store_b128 th:NT.
        __builtin_nontemporal_store(yo, &y[off]);
    }
}

// Scalar fallback path for any non-multiple-of-4 tail (not hit for N=32768,
// but the vector kernel's plane stride is only right when BN % 4 == 0).
__global__ __launch_bounds__(256) void lif_btn_tail_kernel(
    const float* __restrict__ x, float* __restrict__ y, int bn, int start)
{
    const int i = start + blockIdx.x * blockDim.x + threadIdx.x;
    if (i >= bn) return;

    float v = 0.f;
#pragma unroll
    for (int t = 0; t < LIF_T; ++t) {
        const size_t off = (size_t)t * (size_t)bn + (size_t)i;
        const float m = __builtin_fmaf(LIF_TAU, v, __builtin_nontemporal_load(&x[off]));
        const bool f = (m >= LIF_VTH);
        __builtin_nontemporal_store(f ? 1.0f : 0.0f, &y[off]);
        v = f ? 0.0f : m;
    }
}

extern "C" void kernel_launch(void* const* d_in, const int* in_sizes, int n_in,
                              void* d_out, int out_size, void* d_ws, size_t ws_size,
                              hipStream_t stream) {
    (void)n_in; (void)out_size; (void)d_ws; (void)ws_size;

    const float* x = (const float*)d_in[0];
    float* y = (float*)d_out;

    const int total = in_sizes[0];        // T*B*N
    const int bn    = total / LIF_T;      // B*N neurons
    const int bn4   = bn / 4;             // float4 columns
    const int tail0 = bn4 * 4;

    if (bn4 > 0) {
        const int threads = 256;
        const int blocks  = (bn4 + threads - 1) / threads;
        lif_btn_kernel<<<blocks, threads, 0, stream>>>(
            (const v4f*)x, (v4f*)y, bn4);
    }
    if (tail0 < bn) {
        const int threads = 256;
        const int blocks  = (bn - tail0 + threads - 1) / threads;
        lif_btn_tail_kernel<<<blocks, threads, 0, stream>>>(x, y, bn, tail0);
    }
}